// Trainer_83760452206638
// MI455X (gfx1250) — compile-verified
//
#include <hip/hip_runtime.h>
#include <hip/hip_bf16.h>

// Problem dims (match reference)
#define BB 4096
#define SS 64
#define YY 256
#define HH 128
#define EE 32
#define KGp 8
#define KAp 20
#define NC 50
#define NT 5

// Output offsets (floats), concat in reference return order
#define OFF_COUNTRY 0
#define OFF_TYPE   (BB*NC)               // 204800
#define OFF_TASTE  (OFF_TYPE + BB)       // 208896
#define OFF_GPRED  (OFF_TASTE + BB*NT)   // 229376
#define OFF_GTRUE  (OFF_GPRED + BB*KGp)  // 262144
#define OFF_APRED  (OFF_GTRUE + BB*KGp)  // 294912
#define OFF_ATRUE  (OFF_APRED + BB*KAp)  // 376832

// Workspace offsets (floats); total ~7.1 MB
#define WS_X    0                        // pooled X [B, Y]
#define WS_YHID (WS_X + BB*YY)
#define WS_YG   (WS_YHID + BB*HH)
#define WS_YA   (WS_YG + BB*EE)
#define WS_WALL (WS_YA + BB*EE)
#define WS_BALL (WS_WALL + 128*128)

typedef __attribute__((ext_vector_type(2))) float v2f;
typedef __attribute__((ext_vector_type(4))) float v4f;
typedef __attribute__((ext_vector_type(8))) float v8f;

__device__ __forceinline__ float sigmoidf_(float x) { return 1.0f / (1.0f + __expf(-x)); }

// D = A(16x4 f32) * B(4x16 f32) + C(16x16 f32), wave32, fp32 WMMA
__device__ __forceinline__ v8f wmma4(v2f a, v2f b, v8f c) {
  return __builtin_amdgcn_wmma_f32_16x16x4_f32(false, a, false, b, (short)0, c, false, false);
}

// ---------------------------------------------------------------------------
// Max-pool over S: one thread per float4 column of one batch row.
// Streaming 268 MB once -> nontemporal loads (don't churn the 192MB L2).
// 262144 threads keep enough b128 loads in flight to saturate 23.3 TB/s.
// ---------------------------------------------------------------------------
__global__ __launch_bounds__(256)
void k_pool(const float* __restrict__ x_enc, float* __restrict__ X) {
  int tid = blockIdx.x * 256 + threadIdx.x;     // 0 .. B*Y/4-1
  int b  = tid >> 6;                            // Y/4 = 64 float4 per row
  int c4 = tid & 63;
  const v4f* p = (const v4f*)(x_enc + (size_t)b * (SS * YY)) + c4;
  v4f acc = __builtin_nontemporal_load(p);
#pragma unroll 16
  for (int s = 1; s < SS; ++s) {
    v4f v = __builtin_nontemporal_load(p + (size_t)s * (YY / 4));
    acc.x = fmaxf(acc.x, v.x); acc.y = fmaxf(acc.y, v.y);
    acc.z = fmaxf(acc.z, v.z); acc.w = fmaxf(acc.w, v.w);
  }
  ((v4f*)X)[tid] = acc;
}

// ---------------------------------------------------------------------------
// Pack head weights [H,NC]+[H,1]+[H,NT]+[H,E](grape)+[H,E](aroma) -> Wall[128,128]
// ---------------------------------------------------------------------------
__global__ void k_pack(const float* __restrict__ Wcty, const float* __restrict__ bcty,
                       const float* __restrict__ Wtype, const float* __restrict__ btype,
                       const float* __restrict__ Wtaste, const float* __restrict__ btaste,
                       const float* __restrict__ Wgrape, const float* __restrict__ bgrape,
                       const float* __restrict__ Waroma, const float* __restrict__ baroma,
                       float* __restrict__ Wall, float* __restrict__ ball) {
  int tid = blockIdx.x * blockDim.x + threadIdx.x;
  if (tid < 128 * 128) {
    int h = tid >> 7, c = tid & 127;
    float v = 0.f;
    if (c < NC)               v = Wcty[h * NC + c];
    else if (c == NC)         v = Wtype[h];
    else if (c < NC + 1 + NT) v = Wtaste[h * NT + (c - NC - 1)];
    else if (c < 56 + EE)     v = Wgrape[h * EE + (c - 56)];
    else if (c < 88 + EE)     v = Waroma[h * EE + (c - 88)];
    Wall[tid] = v;
  }
  if (tid < 128) {
    int c = tid;
    float v = 0.f;
    if (c < NC)               v = bcty[c];
    else if (c == NC)         v = btype[0];
    else if (c < NC + 1 + NT) v = btaste[c - NC - 1];
    else if (c < 56 + EE)     v = bgrape[c - 56];
    else if (c < 88 + EE)     v = baroma[c - 88];
    ball[c] = v;
  }
}

// ---------------------------------------------------------------------------
// GEMM1: Yhid = elu(X @ W_common + b). Block: 256 thr (8 waves), 16-row M tile;
// wave w owns N-tile w. A tile staged in LDS (padded stride -> conflict-free).
// ---------------------------------------------------------------------------
#define XS_STRIDE 260  // 256 + 4 pad: row m -> bank offset 4*m

__global__ __launch_bounds__(256)
void k_gemm1(const float* __restrict__ X, const float* __restrict__ Wc,
             const float* __restrict__ bc, float* __restrict__ yhid) {
  __shared__ float Xs[16 * XS_STRIDE];
  const int mbase = blockIdx.x * 16;

  // Stage the 16x256 A tile: 1024 float4, 4 per thread, coalesced.
  for (int slot = threadIdx.x; slot < 16 * 64; slot += 256) {
    int m = slot >> 6, c4 = slot & 63;
    v4f v = ((const v4f*)X)[(size_t)(mbase + m) * 64 + c4];
    *(v4f*)(&Xs[m * XS_STRIDE + c4 * 4]) = v;
  }
  __syncthreads();

  const int wave = threadIdx.x >> 5;
  const int lane = threadIdx.x & 31;
  const int half = lane >> 4;   // K-half select per ISA A/B layout
  const int mn   = lane & 15;   // A: row M; B: col N
  const int kb   = half * 2;
  const int ncol = wave * 16 + mn;

  v8f acc = {};
  for (int k0 = 0; k0 < YY; k0 += 4) {
    v2f a, b;
    a.x = Xs[mn * XS_STRIDE + k0 + kb];
    a.y = Xs[mn * XS_STRIDE + k0 + kb + 1];
    b.x = Wc[(k0 + kb) * HH + ncol];       // W_common [Y,H] row-major -> B[k][n]
    b.y = Wc[(k0 + kb + 1) * HH + ncol];
    acc = wmma4(a, b, acc);
  }

  const float bias = bc[ncol];
#pragma unroll
  for (int j = 0; j < 8; ++j) {            // VGPR j holds row M = j + 8*half
    int row = mbase + j + 8 * half;
    float v = acc[j] + bias;
    v = v > 0.f ? v : (__expf(v) - 1.f);   // ELU (alpha=1)
    yhid[(size_t)row * HH + ncol] = v;
  }
}

// ---------------------------------------------------------------------------
// Heads: Z = Yhid[B,128] @ Wall[128,128] + ball, epilogue routes columns.
// A tile staged in LDS once per block (kills 8x redundant global reads).
// ---------------------------------------------------------------------------
#define YS_STRIDE 132  // 128 + 4 pad

__global__ __launch_bounds__(256)
void k_gemm2_heads(const float* __restrict__ yhid, const float* __restrict__ Wall,
                   const float* __restrict__ ball, float* __restrict__ out,
                   float* __restrict__ yg, float* __restrict__ ya) {
  __shared__ float Ys[16 * YS_STRIDE];
  const int mbase = blockIdx.x * 16;

  for (int slot = threadIdx.x; slot < 16 * 32; slot += 256) {  // 512 float4, 2/thread
    int m = slot >> 5, c4 = slot & 31;
    v4f v = ((const v4f*)yhid)[(size_t)(mbase + m) * 32 + c4];
    *(v4f*)(&Ys[m * YS_STRIDE + c4 * 4]) = v;
  }
  __syncthreads();

  const int wave = threadIdx.x >> 5;
  const int lane = threadIdx.x & 31;
  const int half = lane >> 4;
  const int mn   = lane & 15;
  const int kb   = half * 2;
  const int col  = wave * 16 + mn;

  v8f acc = {};
  for (int k0 = 0; k0 < HH; k0 += 4) {
    v2f a, b;
    a.x = Ys[mn * YS_STRIDE + k0 + kb];
    a.y = Ys[mn * YS_STRIDE + k0 + kb + 1];
    b.x = Wall[(k0 + kb) * 128 + col];
    b.y = Wall[(k0 + kb + 1) * 128 + col];
    acc = wmma4(a, b, acc);
  }

  const float bias = ball[col];
#pragma unroll
  for (int j = 0; j < 8; ++j) {
    int row = mbase + j + 8 * half;
    float v = acc[j] + bias;
    if (col < NC)                 out[OFF_COUNTRY + (size_t)row * NC + col] = v;
    else if (col == NC)           out[OFF_TYPE + row] = sigmoidf_(v);
    else if (col < NC + 1 + NT)   out[OFF_TASTE + (size_t)row * NT + (col - NC - 1)] = sigmoidf_(v);
    else if (col < 56 + EE)       yg[(size_t)row * EE + (col - 56)] = v;   // y_grape
    else if (col < 88 + EE)       ya[(size_t)row * EE + (col - 88)] = v;   // y_aroma
    // cols 120..127: padding, discard
  }
}

// ---------------------------------------------------------------------------
// Ragged gathers: dot(y_head[b], emb[idx]) instead of full [B,V] scoring.
// ---------------------------------------------------------------------------
__global__ void k_grapes(const int* __restrict__ grapes, const float* __restrict__ scales,
                         const float* __restrict__ yg, const float* __restrict__ emb,
                         float* __restrict__ out) {
  int tid = blockIdx.x * blockDim.x + threadIdx.x;
  if (tid >= BB * KGp) return;
  int b = tid / KGp;
  int g = grapes[tid];
  const v4f* yv = (const v4f*)(yg + (size_t)b * EE);
  const v4f* ev = (const v4f*)(emb + (size_t)g * EE);
  float dot = 0.f;
#pragma unroll
  for (int j = 0; j < EE / 4; ++j) {
    v4f u = yv[j], w = ev[j];
    dot += u.x * w.x + u.y * w.y + u.z * w.z + u.w * w.w;
  }
  out[OFF_GPRED + tid] = (g != 0) ? sigmoidf_(dot) : 0.f;   // mask = (grapes != 0)
  out[OFF_GTRUE + tid] = scales[tid];                       // s * (s!=0) == s
}

__global__ void k_aromas(const int* __restrict__ aromas, const float* __restrict__ scales,
                         const float* __restrict__ ya, const float* __restrict__ emb,
                         float* __restrict__ out) {
  int tid = blockIdx.x * blockDim.x + threadIdx.x;
  if (tid >= BB * KAp) return;
  int b = tid / KAp;
  int a = aromas[tid];
  float s = scales[tid];
  const v4f* yv = (const v4f*)(ya + (size_t)b * EE);
  const v4f* ev = (const v4f*)(emb + (size_t)a * EE);
  float dot = 0.f;
#pragma unroll
  for (int j = 0; j < EE / 4; ++j) {
    v4f u = yv[j], w = ev[j];
    dot += u.x * w.x + u.y * w.y + u.z * w.z + u.w * w.w;
  }
  out[OFF_APRED + tid] = (s != 0.f) ? sigmoidf_(dot) : 0.f; // mask = (scales != 0)
  out[OFF_ATRUE + tid] = s;
}

// ---------------------------------------------------------------------------
extern "C" void kernel_launch(void* const* d_in, const int* in_sizes, int n_in,
                              void* d_out, int out_size, void* d_ws, size_t ws_size,
                              hipStream_t stream) {
  const float* x_enc  = (const float*)d_in[0];
  const int*   grapes = (const int*)  d_in[1];
  const float* gscal  = (const float*)d_in[2];
  const int*   aromas = (const int*)  d_in[3];
  const float* ascal  = (const float*)d_in[4];
  const float* Wc     = (const float*)d_in[5];
  const float* bc     = (const float*)d_in[6];
  const float* Wcty   = (const float*)d_in[7];
  const float* bcty   = (const float*)d_in[8];
  const float* Wtype  = (const float*)d_in[9];
  const float* btype  = (const float*)d_in[10];
  const float* Wtaste = (const float*)d_in[11];
  const float* btaste = (const float*)d_in[12];
  const float* Waroma = (const float*)d_in[13];
  const float* baroma = (const float*)d_in[14];
  const float* Wgrape = (const float*)d_in[15];
  const float* bgrape = (const float*)d_in[16];
  const float* gemb   = (const float*)d_in[17];
  const float* aemb   = (const float*)d_in[18];

  float* out  = (float*)d_out;
  float* ws   = (float*)d_ws;
  float* X    = ws + WS_X;
  float* yhid = ws + WS_YHID;
  float* yg   = ws + WS_YG;
  float* ya   = ws + WS_YA;
  float* Wall = ws + WS_WALL;
  float* ball = ws + WS_BALL;

  k_pack<<<(128 * 128 + 255) / 256, 256, 0, stream>>>(
      Wcty, bcty, Wtype, btype, Wtaste, btaste, Wgrape, bgrape, Waroma, baroma, Wall, ball);
  k_pool<<<(BB * YY / 4) / 256, 256, 0, stream>>>(x_enc, X);
  k_gemm1<<<BB / 16, 256, 0, stream>>>(X, Wc, bc, yhid);
  k_gemm2_heads<<<BB / 16, 256, 0, stream>>>(yhid, Wall, ball, out, yg, ya);
  k_grapes<<<(BB * KGp + 255) / 256, 256, 0, stream>>>(grapes, gscal, yg, gemb, out);
  k_aromas<<<(BB * KAp + 255) / 256, 256, 0, stream>>>(aromas, ascal, ya, aemb, out);
}